// EURNNCell_15693810499971
// MI455X (gfx1250) — compile-verified
//
#include <hip/hip_runtime.h>

typedef float v2f __attribute__((ext_vector_type(2)));
typedef float v8f __attribute__((ext_vector_type(8)));

#define MDIM 8192
#define NDIM 1024
#define KDIM 1024
#define LHALF 16        // L/2 scan steps

#define BM 128
#define BN 128
#define BK 32
#define SA (4 * BM + 8)  // slab stride (floats) per k-group of 4; +8 keeps banks spread
#define SB (4 * BN + 8)

// ---------------------------------------------------------------------------
// K1: rotation coefficient tables (faithful to torch stack+raw-reshape order)
// dA/oA/dB/oB each [16][1024] floats in d_ws.
// ---------------------------------------------------------------------------
__global__ __launch_bounds__(256)
void eunn_coeff_kernel(const float* __restrict__ thetaA,   // (512,16)
                       const float* __restrict__ thetaB,   // (511,16)
                       float* __restrict__ dA, float* __restrict__ oA,
                       float* __restrict__ dB, float* __restrict__ oB)
{
    int idx = blockIdx.x * blockDim.x + threadIdx.x;       // 0..16383
    if (idx >= LHALF * NDIM) return;
    const int t = idx >> 10;
    const int j = idx & 1023;

    // A-tables: flat = t*1024 + j == idx ; (512,16,2) raw reshape -> (16,1024)
    {
        const int a = idx >> 5;
        const int b = (idx & 31) >> 1;
        const int c = idx & 1;
        const float th = thetaA[a * 16 + b];
        dA[idx] = cosf(th);
        oA[idx] = (c ? 1.0f : -1.0f) * sinf(th);
    }
    // B-tables: interior from (511,16,2) raw reshape -> (16,1022); endpoints 1/0
    if (j == 0 || j == 1023) {
        dB[idx] = 1.0f;
        oB[idx] = 0.0f;
    } else {
        const int f = t * 1022 + (j - 1);
        const int a = f >> 5;
        const int b = (f & 31) >> 1;
        const int c = f & 1;
        const float th = thetaB[a * 16 + b];
        dB[idx] = cosf(th);
        oB[idx] = (c ? 1.0f : -1.0f) * sinf(th);
    }
}

// ---------------------------------------------------------------------------
// K2: h = input_ @ U + hx  via V_WMMA_F32_16X16X4_F32 (native fp32 matrix op)
// Block: 256 thr = 8 waves (4x2), wave tile 32x64 = 2x4 WMMA accumulators.
// LDS is fragment-ready: [k/4][row][4] so each lane's operand pair (k+koff,
// k+koff+1) is one contiguous ds_load_b64 straight into an even VGPR pair.
// ---------------------------------------------------------------------------
__global__ __launch_bounds__(256)
void gemm_hx_kernel(const float* __restrict__ A,    // input_ (M x K)
                    const float* __restrict__ Bg,   // U      (K x N)
                    const float* __restrict__ hx,   // (M x N)
                    float* __restrict__ out)        // h      (M x N)
{
    __shared__ __align__(16) float AsF[(BK / 4) * SA];
    __shared__ __align__(16) float BsF[(BK / 4) * SB];

    const int tid    = threadIdx.x;
    const int lane   = tid & 31;            // wave32
    const int wave   = tid >> 5;            // 0..7
    const int wm     = wave & 3;            // M direction 0..3
    const int wn     = wave >> 2;           // N direction 0..1
    const int blockM = blockIdx.x * BM;
    const int blockN = blockIdx.y * BN;

    const int lane15 = lane & 15;
    const int koff   = (lane >> 4) << 1;    // lanes 16..31 hold K+2,K+3 (32-bit A/B layout)

    v8f acc[2][4] = {};                     // 2x4 tiles of 16x16 f32

    // A loader: thread -> (m = tid/8 + 32p, kgrp = tid&7); float4 along K.
    const int a_m0 = tid >> 3;              // 0..31
    const int a_kg = tid & 7;               // 0..7 (k-group of 4)
    // B loader: thread -> (n = tid&127, k0 = (tid>>7)*4 + 8p); scalar along K,
    // coalesced across lanes in N; one b128 LDS store per k-group.
    const int b_n  = tid & 127;
    const int b_kg0 = (tid >> 7);           // 0..1

    const int mA = wm * 32 + lane15;
    const int nB = wn * 64 + lane15;

    for (int kb = 0; kb < KDIM; kb += BK) {
        __syncthreads();                    // protect LDS from previous readers
        // ---- fill A (fragment-ready) ----
        #pragma unroll
        for (int p = 0; p < 4; ++p) {
            const int m = a_m0 + 32 * p;
            const float4 v = *(const float4*)(A + (size_t)(blockM + m) * KDIM + kb + a_kg * 4);
            *(float4*)(AsF + a_kg * SA + m * 4) = v;
        }
        // ---- fill B (fragment-ready) ----
        #pragma unroll
        for (int p = 0; p < 4; ++p) {
            const int kg = b_kg0 + 2 * p;   // 0..7
            const int k  = kb + kg * 4;
            float4 v;
            v.x = Bg[(size_t)(k + 0) * NDIM + blockN + b_n];
            v.y = Bg[(size_t)(k + 1) * NDIM + blockN + b_n];
            v.z = Bg[(size_t)(k + 2) * NDIM + blockN + b_n];
            v.w = Bg[(size_t)(k + 3) * NDIM + blockN + b_n];
            *(float4*)(BsF + kg * SB + b_n * 4) = v;
        }
        __syncthreads();

        if (kb + BK < KDIM) {               // global_prefetch_b8 of next K tiles
            __builtin_prefetch(A + (size_t)(blockM + a_m0) * KDIM + kb + BK + a_kg * 4, 0, 0);
            __builtin_prefetch(Bg + (size_t)(kb + BK + b_kg0 * 4) * NDIM + blockN + b_n, 0, 0);
        }

        // ---- 8 k-groups x 8 WMMAs ----
        #pragma unroll
        for (int kg = 0; kg < BK / 4; ++kg) {
            const float* aB = AsF + kg * SA + koff;
            const float* bB = BsF + kg * SB + koff;
            v2f a0 = *(const v2f*)(aB + (mA     ) * 4);
            v2f a1 = *(const v2f*)(aB + (mA + 16) * 4);
            v2f b0 = *(const v2f*)(bB + (nB     ) * 4);
            v2f b1 = *(const v2f*)(bB + (nB + 16) * 4);
            v2f b2 = *(const v2f*)(bB + (nB + 32) * 4);
            v2f b3 = *(const v2f*)(bB + (nB + 48) * 4);
            acc[0][0] = __builtin_amdgcn_wmma_f32_16x16x4_f32(false, a0, false, b0,
                                                              (short)0, acc[0][0], false, false);
            acc[0][1] = __builtin_amdgcn_wmma_f32_16x16x4_f32(false, a0, false, b1,
                                                              (short)0, acc[0][1], false, false);
            acc[0][2] = __builtin_amdgcn_wmma_f32_16x16x4_f32(false, a0, false, b2,
                                                              (short)0, acc[0][2], false, false);
            acc[0][3] = __builtin_amdgcn_wmma_f32_16x16x4_f32(false, a0, false, b3,
                                                              (short)0, acc[0][3], false, false);
            acc[1][0] = __builtin_amdgcn_wmma_f32_16x16x4_f32(false, a1, false, b0,
                                                              (short)0, acc[1][0], false, false);
            acc[1][1] = __builtin_amdgcn_wmma_f32_16x16x4_f32(false, a1, false, b1,
                                                              (short)0, acc[1][1], false, false);
            acc[1][2] = __builtin_amdgcn_wmma_f32_16x16x4_f32(false, a1, false, b2,
                                                              (short)0, acc[1][2], false, false);
            acc[1][3] = __builtin_amdgcn_wmma_f32_16x16x4_f32(false, a1, false, b3,
                                                              (short)0, acc[1][3], false, false);
        }
    }

    // Epilogue: D-layout is VGPR v,lane l -> (M = v + 8*(l>=16), N = l&15). Fuse +hx.
    const int hi = (lane >> 4) << 3;        // 0 or 8
    #pragma unroll
    for (int ti = 0; ti < 2; ++ti) {
        #pragma unroll
        for (int tj = 0; tj < 4; ++tj) {
            const int mBase = blockM + wm * 32 + ti * 16 + hi;
            const int nCol  = blockN + wn * 64 + tj * 16 + lane15;
            #pragma unroll
            for (int v = 0; v < 8; ++v) {
                const size_t idx = (size_t)(mBase + v) * NDIM + nCol;
                out[idx] = acc[ti][tj][v] + hx[idx];
            }
        }
    }
}

// ---------------------------------------------------------------------------
// K3: EUNN scan (16 steps x 2 rotation stages) + modReLU, in place on d_out.
// One workgroup per batch row; row lives 4 floats/thread in registers,
// stage-B gather goes through a 4KB LDS row.
// ---------------------------------------------------------------------------
__global__ __launch_bounds__(256)
void eunn_modrelu_kernel(float* __restrict__ h,        // (B x N), in/out
                         const float* __restrict__ dA,
                         const float* __restrict__ oA,
                         const float* __restrict__ dB,
                         const float* __restrict__ oB,
                         const float* __restrict__ bias)
{
    __shared__ float xs[NDIM];
    const int row = blockIdx.x;
    const int tid = threadIdx.x;
    const int j0  = tid * 4;
    float* __restrict__ hrow = h + (size_t)row * NDIM;

    float4 x = *(const float4*)(hrow + j0);

    for (int t = 0; t < LHALF; ++t) {
        const float4 da = *(const float4*)(dA + t * NDIM + j0);
        const float4 oa = *(const float4*)(oA + t * NDIM + j0);
        // stage A: swap within adjacent pairs -> register-local
        float4 xa;
        xa.x = x.x * da.x + x.y * oa.x;
        xa.y = x.y * da.y + x.x * oa.y;
        xa.z = x.z * da.z + x.w * oa.z;
        xa.w = x.w * da.w + x.z * oa.w;

        *(float4*)(xs + j0) = xa;
        __syncthreads();

        const float4 db = *(const float4*)(dB + t * NDIM + j0);
        const float4 ob = *(const float4*)(oB + t * NDIM + j0);
        float y[4];
        #pragma unroll
        for (int u = 0; u < 4; ++u) {
            const int j   = j0 + u;
            const int src = (j < 512) ? (2 * j) : (2 * j - 1023);  // endpoints fall out
            y[u] = xs[src];
        }
        x.x = xa.x * db.x + y[0] * ob.x;
        x.y = xa.y * db.y + y[1] * ob.y;
        x.z = xa.z * db.z + y[2] * ob.z;
        x.w = xa.w * db.w + y[3] * ob.w;
        __syncthreads();                    // before next iter overwrites xs
    }

    // modReLU: sign(h) * relu(|h| + bias)
    const float4 bb = *(const float4*)(bias + j0);
    float4 o;
    {
        float s;
        s   = (x.x > 0.f) ? 1.f : ((x.x < 0.f) ? -1.f : 0.f);
        o.x = s * fmaxf(fabsf(x.x) + bb.x, 0.f);
        s   = (x.y > 0.f) ? 1.f : ((x.y < 0.f) ? -1.f : 0.f);
        o.y = s * fmaxf(fabsf(x.y) + bb.y, 0.f);
        s   = (x.z > 0.f) ? 1.f : ((x.z < 0.f) ? -1.f : 0.f);
        o.z = s * fmaxf(fabsf(x.z) + bb.z, 0.f);
        s   = (x.w > 0.f) ? 1.f : ((x.w < 0.f) ? -1.f : 0.f);
        o.w = s * fmaxf(fabsf(x.w) + bb.w, 0.f);
    }
    *(float4*)(hrow + j0) = o;
}

// ---------------------------------------------------------------------------
extern "C" void kernel_launch(void* const* d_in, const int* in_sizes, int n_in,
                              void* d_out, int out_size, void* d_ws, size_t ws_size,
                              hipStream_t stream)
{
    const float* input_ = (const float*)d_in[0];   // (8192,1024)
    const float* hx     = (const float*)d_in[1];   // (8192,1024)
    const float* U      = (const float*)d_in[2];   // (1024,1024)
    const float* thetaA = (const float*)d_in[3];   // (512,16)
    const float* thetaB = (const float*)d_in[4];   // (511,16)
    const float* bias   = (const float*)d_in[5];   // (1024,)
    // d_in[6] == i, unused by the reference math
    float* out = (float*)d_out;

    float* ws = (float*)d_ws;                      // 256KB of coeff tables
    float* dA = ws;
    float* oA = ws + 16384;
    float* dB = ws + 32768;
    float* oB = ws + 49152;

    eunn_coeff_kernel<<<(LHALF * NDIM + 255) / 256, 256, 0, stream>>>(
        thetaA, thetaB, dA, oA, dB, oB);

    dim3 grid(MDIM / BM, NDIM / BN);
    gemm_hx_kernel<<<grid, 256, 0, stream>>>(input_, U, hx, out);

    eunn_modrelu_kernel<<<MDIM, 256, 0, stream>>>(out, dA, oA, dB, oB, bias);
}